// ModuleCorrelation_26611617366061
// MI455X (gfx1250) — compile-verified
//
#include <hip/hip_runtime.h>

typedef __attribute__((ext_vector_type(2))) float v2f;
typedef __attribute__((ext_vector_type(8))) float v8f;

#define Bn 8
#define Cn 128
#define Hn 128
#define Wn 256
#define NDISP 81
#define HWn (Hn * Wn)
#define CHWn (Cn * HWn)

// One wave32 handles one (b, y, 16-wide x tile). For each of 9 dy values it
// computes two 16x16xK=128 fp32 GEMMs:
//   A[m,k]  = first [b, k, y,     x0+m]
//   B1[k,n] = second[b, k, y+dy,  x0-8+n]   (zero outside the image)
//   B2[k,n] = second[b, k, y+dy,  x0+8+n]
// O[m,n] = corr at (x0+m) with dx = (x1+n)-(x0+m); dx in [-4,4] is written.
// Tile1 covers m+dx<=7, tile2 covers m+dx>=8 -> each output written once.
//
// Zero padding is handled WITHOUT load predication: B is loaded from clamped
// (always-valid) addresses, and the padding mask is applied at store time
// (an OOB B column/row zeroes the whole dot product, so out = 0 there).
__global__ __launch_bounds__(256) void corr_wmma_f32(
    const float* __restrict__ first,
    const float* __restrict__ second,
    float* __restrict__ out)
{
    const int lane = threadIdx.x & 31;
    const int wave = threadIdx.x >> 5;
    const int tile = blockIdx.x * 8 + wave;        // 16384 tiles total

    const int xt = tile & 15;                      // W/16 = 16 tiles in x
    const int y  = (tile >> 4) & (Hn - 1);
    const int b  = tile >> 11;                     // b slowest -> L2 locality
    const int x0 = xt << 4;

    const int n  = lane & 15;                      // M for A loads, N for B/C
    const int hi = lane >> 4;                      // upper half-wave
    const int ch0 = hi * 2;                        // K offset within 4-chunk

    // ---- Load A fragments (resident across the dy loop): 64 VGPRs ----
    const float* fb = first + (size_t)b * CHWn + (size_t)y * Wn + x0 + n;
    float a[64];
#pragma unroll
    for (int ks = 0; ks < 32; ++ks) {
        const float* fp = fb + (size_t)(4 * ks + ch0) * HWn;
        a[2 * ks + 0] = fp[0];
        a[2 * ks + 1] = fp[HWn];
    }

    const int xb1 = x0 - 8 + n;
    const int xb2 = x0 + 8 + n;
    const bool x1ok = (unsigned)xb1 < (unsigned)Wn;
    const bool x2ok = (unsigned)xb2 < (unsigned)Wn;
    const int xb1c = min(max(xb1, 0), Wn - 1);     // clamped (always valid)
    const int xb2c = min(max(xb2, 0), Wn - 1);
    const float* sb = second + (size_t)b * CHWn + (size_t)ch0 * HWn;
    float* ob = out + (size_t)b * NDISP * HWn + (size_t)y * Wn + x0;

#pragma unroll 1
    for (int dyi = 0; dyi < 9; ++dyi) {
        const int ys = y + dyi - 4;
        const bool rok = (unsigned)ys < (unsigned)Hn;
        const bool l1 = rok && x1ok;
        const bool l2 = rok && x2ok;
        const int ysc = min(max(ys, 0), Hn - 1);   // clamped (always valid)
        const float* s1 = sb + (size_t)ysc * Wn + xb1c;
        const float* s2 = sb + (size_t)ysc * Wn + xb2c;

        v8f acc1 = {0.f, 0.f, 0.f, 0.f, 0.f, 0.f, 0.f, 0.f};
        v8f acc2 = {0.f, 0.f, 0.f, 0.f, 0.f, 0.f, 0.f, 0.f};

#pragma unroll
        for (int ks = 0; ks < 32; ++ks) {
            // Unconditional loads: schedulable, clause-able, no exec churn.
            const float b10 = s1[(size_t)(4 * ks) * HWn];
            const float b11 = s1[(size_t)(4 * ks + 1) * HWn];
            const float b20 = s2[(size_t)(4 * ks) * HWn];
            const float b21 = s2[(size_t)(4 * ks + 1) * HWn];
            v2f av  = {a[2 * ks], a[2 * ks + 1]};
            v2f bv1 = {b10, b11};
            v2f bv2 = {b20, b21};
            acc1 = __builtin_amdgcn_wmma_f32_16x16x4_f32(
                false, av, false, bv1, (short)0, acc1, false, false);
            acc2 = __builtin_amdgcn_wmma_f32_16x16x4_f32(
                false, av, false, bv2, (short)0, acc2, false, false);
        }

        // ---- Scatter valid (m, dx) entries; padding mask applied here ----
#pragma unroll
        for (int v = 0; v < 8; ++v) {
            const int m  = v + hi * 8;
            const int d1 = n - m - 8;   // dx from tile1
            const int d2 = n - m + 8;   // dx from tile2
            if (d1 >= -4 && d1 <= 4)
                ob[(size_t)(dyi * 9 + d1 + 4) * HWn + m] =
                    l1 ? acc1[v] * 0.0078125f : 0.0f;
            if (d2 >= -4 && d2 <= 4)
                ob[(size_t)(dyi * 9 + d2 + 4) * HWn + m] =
                    l2 ? acc2[v] * 0.0078125f : 0.0f;
        }
    }
}

extern "C" void kernel_launch(void* const* d_in, const int* in_sizes, int n_in,
                              void* d_out, int out_size, void* d_ws, size_t ws_size,
                              hipStream_t stream) {
    const float* first  = (const float*)d_in[0];
    const float* second = (const float*)d_in[1];
    float* out = (float*)d_out;

    // 8 * 128 * 16 = 16384 waves, 8 waves (256 threads) per block
    const int total_tiles = Bn * Hn * (Wn / 16);
    dim3 grid(total_tiles / 8);
    dim3 block(256);
    corr_wmma_f32<<<grid, block, 0, stream>>>(first, second, out);
}